// CapsuleLayer_70033736729248
// MI455X (gfx1250) — compile-verified
//
#include <hip/hip_runtime.h>
#include <math.h>

typedef __attribute__((ext_vector_type(2))) float v2f;
typedef __attribute__((ext_vector_type(8))) float v8f;

#define M_DIM   32
#define K_DIM   512
#define N_DIM   16384
#define NCAPS   256
#define ODIM    64

// ---------------------------------------------------------------------------
// GEMM: h[32,16384] = x[32,512] @ W[512,16384], fp32 via V_WMMA_F32_16X16X4_F32
// Wave tiling: each wave owns one 16-row M tile and 64 columns (4 WMMA tiles).
// ---------------------------------------------------------------------------
__global__ __launch_bounds__(256) void caps_gemm_f32wmma(
    const float* __restrict__ x, const float* __restrict__ W,
    float* __restrict__ h) {
  __shared__ float xs[M_DIM * K_DIM];  // 64 KB: entire x staged in LDS

  const int tid = threadIdx.x;
  {
    const float4* src = (const float4*)x;
    float4* dst = (float4*)xs;
#pragma unroll
    for (int i = 0; i < (M_DIM * K_DIM / 4) / 256; ++i)
      dst[tid + i * 256] = src[tid + i * 256];
  }
  __syncthreads();

  const int lane = tid & 31;
  const int wave = (blockIdx.x << 3) | (tid >> 5);  // 0..511
  const int mTile = wave & 1;
  const int nGroup = wave >> 1;                     // 0..255
  const int m0 = mTile * 16;
  const int n0 = nGroup * 64;

  const int half = lane >> 4;  // 0: K+{0,1}, 1: K+{2,3}  (ISA A-layout 16x4)
  const int lr   = lane & 15;

  const float* aPtr = &xs[(m0 + lr) * K_DIM + 2 * half];
  const float* bPtr = W + (size_t)(2 * half) * N_DIM + (n0 + lr);

  v8f acc0 = {}, acc1 = {}, acc2 = {}, acc3 = {};

#pragma unroll 2
  for (int k = 0; k < K_DIM; k += 4) {
    // A fragment from LDS (float2 -> ds_load_b64), reused for 4 N tiles
    v2f A;
    A.x = aPtr[0];
    A.y = aPtr[1];
    // B fragments: B[row=k+2*half..+1][col], lanes carry N
    v2f B0, B1, B2, B3;
    B0.x = bPtr[0];  B0.y = bPtr[N_DIM + 0];
    B1.x = bPtr[16]; B1.y = bPtr[N_DIM + 16];
    B2.x = bPtr[32]; B2.y = bPtr[N_DIM + 32];
    B3.x = bPtr[48]; B3.y = bPtr[N_DIM + 48];
    // stream W ahead (global_prefetch_b8)
    __builtin_prefetch(bPtr + (size_t)16 * N_DIM, 0, 1);

    acc0 = __builtin_amdgcn_wmma_f32_16x16x4_f32(false, A, false, B0, (short)0,
                                                 acc0, false, false);
    acc1 = __builtin_amdgcn_wmma_f32_16x16x4_f32(false, A, false, B1, (short)0,
                                                 acc1, false, false);
    acc2 = __builtin_amdgcn_wmma_f32_16x16x4_f32(false, A, false, B2, (short)0,
                                                 acc2, false, false);
    acc3 = __builtin_amdgcn_wmma_f32_16x16x4_f32(false, A, false, B3, (short)0,
                                                 acc3, false, false);
    aPtr += 4;
    bPtr += (size_t)4 * N_DIM;
  }

  // D layout (ISA 16x16 f32 C/D): VGPR r -> row m0 + 8*half + r, col = lane
#pragma unroll
  for (int r = 0; r < 8; ++r) {
    float* dp = h + (size_t)(m0 + half * 8 + r) * N_DIM + n0 + lr;
    dp[0]  = acc0[r];
    dp[16] = acc1[r];
    dp[32] = acc2[r];
    dp[48] = acc3[r];
  }
}

// ---------------------------------------------------------------------------
// Dynamic routing (3 iterations), literal softmax-sum scaling.
// One wave32 per (batch b, out-dim d); 8 capsule values per lane in registers.
// ---------------------------------------------------------------------------
__global__ __launch_bounds__(256) void caps_route3(float* __restrict__ h) {
  const int lane = threadIdx.x & 31;
  const int wave = (blockIdx.x * blockDim.x + threadIdx.x) >> 5;  // 0..2047
  const int b = wave >> 6;   // batch
  const int d = wave & 63;   // out-dim

  float* hp = h + (size_t)b * N_DIM + d;
  float v[8];
#pragma unroll
  for (int r = 0; r < 8; ++r)
    v[r] = hp[(size_t)(lane + 32 * r) * ODIM];

#pragma unroll
  for (int t = 0; t < 3; ++t) {
    // max over 256 capsules
    float m = v[0];
#pragma unroll
    for (int r = 1; r < 8; ++r) m = fmaxf(m, v[r]);
#pragma unroll
    for (int off = 16; off > 0; off >>= 1)
      m = fmaxf(m, __shfl_xor(m, off, 32));
    // sum of exp
    float e[8];
    float se = 0.f;
#pragma unroll
    for (int r = 0; r < 8; ++r) { e[r] = expf(v[r] - m); se += e[r]; }
#pragma unroll
    for (int off = 16; off > 0; off >>= 1)
      se += __shfl_xor(se, off, 32);
    // sum of softmax values c_i = e_i / S (literal, like the reference)
    float cs = 0.f;
#pragma unroll
    for (int r = 0; r < 8; ++r) cs += e[r] / se;
#pragma unroll
    for (int off = 16; off > 0; off >>= 1)
      cs += __shfl_xor(cs, off, 32);
    // h[b,j,d] *= sum_i c[b,i,d]
#pragma unroll
    for (int r = 0; r < 8; ++r) v[r] *= cs;
  }

#pragma unroll
  for (int r = 0; r < 8; ++r)
    hp[(size_t)(lane + 32 * r) * ODIM] = v[r];
}

extern "C" void kernel_launch(void* const* d_in, const int* in_sizes, int n_in,
                              void* d_out, int out_size, void* d_ws,
                              size_t ws_size, hipStream_t stream) {
  const float* x = (const float*)d_in[0];     // [32, 512]
  const float* W = (const float*)d_in[1];     // [1, 512, 16384] -> [512,16384]
  float* out = (float*)d_out;                 // [32, 256, 64] = [32, 16384]

  // 512 waves of GEMM work: 64 blocks x 8 waves
  caps_gemm_f32wmma<<<64, 256, 0, stream>>>(x, W, out);
  // 2048 routing waves: 256 blocks x 8 waves (in-place on d_out)
  caps_route3<<<256, 256, 0, stream>>>(out);
}